// PatchSelect_48034914238599
// MI455X (gfx1250) — compile-verified
//
#include <hip/hip_runtime.h>
#include <hip/hip_bf16.h>
#include <stdint.h>

typedef __attribute__((ext_vector_type(2))) float v2f;
typedef __attribute__((ext_vector_type(8))) float v8f;

#define NB    8        // batches
#define CK    12288    // 3*64*64 elements per lr image
#define REFB  27648    // 3*96*96 elements per ref image
#define OO    33       // positions per axis (96-64+1)
#define NPOS  1089     // 33*33
#define WREF  96
#define CHST  9216     // 96*96 channel stride in ref
#define NSLICE 8       // CK slices (one per wave)
#define ROWS_PER_SLICE 24         // 1536 elements = 24 rows of 64

// ---------------------------------------------------------------------------
// Pass 1: per-(batch, 16-position group, CK-slice) partial SADs via WMMA.
//   D = A(ones 16x4) x B(|diff| 4x16) + C  => C accumulates exact f32 column
//   sums: C[r,c] = SAD of position c over this wave's CK slice.
// Element -> K-row mapping is arbitrary (A is all ones), so lane (h,c) feeds
// the CONTIGUOUS pair (4t+2h, 4t+2h+1) per chunk: b64 global + b64 LDS loads,
// with row-relative immediate offsets (no per-chunk address VALU).
// lr[b] staged in LDS with gfx1250 async global->LDS DMA (ASYNCcnt path).
// ---------------------------------------------------------------------------
__global__ __launch_bounds__(256) void ps_sad(const float* __restrict__ lr,
                                              const float* __restrict__ ref,
                                              float* __restrict__ part) {
  __shared__ __align__(16) float s_lr[CK];   // 48 KB

  const int tid     = threadIdx.x;
  const int b       = blockIdx.x;            // 0..7
  const int grp     = blockIdx.y;            // 0..98 = oy*3 + gx
  const int oy      = grp / 3;
  const int ox_base = (grp % 3) * 16;

  // ---- async global -> LDS stage of lr[b] (12288 floats, b128 per lane) ----
  {
    const uint64_t gbase = (uint64_t)(uintptr_t)(lr + (size_t)b * CK);
    const uint32_t lbase = (uint32_t)(uintptr_t)(&s_lr[0]);
#pragma unroll
    for (int i = 0; i < 12; ++i) {
      const uint32_t off = (uint32_t)(i * 256 + tid) * 16u;
      const uint64_t ga  = gbase + off;
      const uint32_t la  = lbase + off;
      asm volatile("global_load_async_to_lds_b128 %0, %1, off"
                   :: "v"(la), "v"(ga) : "memory");
    }
    asm volatile("s_wait_asynccnt 0" ::: "memory");
  }
  __syncthreads();

  const int lane = tid & 31;
  const int wave = tid >> 5;                 // CK slice id, 0..7
  const int h    = (lane >> 4) & 1;          // which element pair of the chunk
  const int c    = lane & 15;                // B-matrix column = position
  const int ox_raw = ox_base + c;
  const int ox_eff = ox_raw <= 32 ? ox_raw : 32;   // clamp for safe addressing
  const bool valid = (lane < 16) && (ox_raw <= 32);

  // ref pointer pre-offset by (batch, oy row, ox column, lane pair)
  const float* __restrict__ refp =
      ref + (size_t)b * REFB + oy * WREF + ox_eff + 2 * h;

  const v2f ones = {1.0f, 1.0f};
  v8f acc = {0.f, 0.f, 0.f, 0.f, 0.f, 0.f, 0.f, 0.f};

  const int rbeg = wave * ROWS_PER_SLICE;    // first lr/ref row of this slice
  for (int r = 0; r < ROWS_PER_SLICE; ++r) {
    const int rr = rbeg + r;                 // global row 0..191
    const int ch = rr >> 6;                  // channel 0..2
    const int y  = rr & 63;                  // row within channel
    const float* __restrict__ rowg = refp + ch * CHST + y * WREF;
    const float* __restrict__ rowl = &s_lr[rr * 64 + 2 * h];
#pragma unroll
    for (int t = 0; t < 16; ++t) {           // 16 chunks of 4 elements per row
      // contiguous pair -> single b64 loads, immediate offsets after unroll
      const v2f g = *(const v2f*)(rowg + 4 * t);      // ref (4B-aligned ok)
      const v2f l = *(const v2f*)(rowl + 4 * t);      // lr from LDS (8B align)
      v2f bm;
      bm.x = fabsf(g.x - l.x);
      bm.y = fabsf(g.y - l.y);
      // D = ones(16x4) * bm(4x16) + acc : exact f32 column sums
      acc = __builtin_amdgcn_wmma_f32_16x16x4_f32(false, ones, false, bm,
                                                  (short)0, acc, false, false);
    }
  }

  // acc[0] (lane<16) = D[0, c] = SAD(position ox_raw) over this slice.
  // Deterministic writeout: one unique (b,pos,slice) slot per valid lane.
  if (valid) {
    const int pos = oy * OO + ox_raw;
    part[((size_t)b * NPOS + pos) * NSLICE + wave] = acc[0];
  }
}

// ---------------------------------------------------------------------------
// Pass 2: per-batch deterministic sum of slice partials + packed-key argmin.
// key = (f32bits(sum) << 32) | index  (sums >= 0 so IEEE bits are monotone);
// min(key) => min value with first-index tie-break, matching jnp.argmin.
// ---------------------------------------------------------------------------
__global__ __launch_bounds__(256) void ps_argmin(const float* __restrict__ part,
                                                 float* __restrict__ out) {
  __shared__ unsigned long long red[256];
  const int b = blockIdx.x;

  unsigned long long best = ~0ull;
  for (int i = threadIdx.x; i < NPOS; i += 256) {
    const float* p = part + ((size_t)b * NPOS + i) * NSLICE;
    float s = 0.f;
#pragma unroll
    for (int w = 0; w < NSLICE; ++w) s += p[w];   // fixed order: deterministic
    const unsigned long long key =
        ((unsigned long long)__float_as_uint(s) << 32) | (unsigned)i;
    best = (key < best) ? key : best;
  }
  red[threadIdx.x] = best;
  __syncthreads();
  for (int off = 128; off > 0; off >>= 1) {
    if (threadIdx.x < off) {
      const unsigned long long o = red[threadIdx.x + off];
      if (o < red[threadIdx.x]) red[threadIdx.x] = o;
    }
    __syncthreads();
  }
  if (threadIdx.x == 0) {
    const unsigned long long k = red[0];
    out[b]     = (float)(unsigned)(k & 0xFFFFFFFFull);            // hard index
    out[9 + b] = __uint_as_float((unsigned)(k >> 32)) *
                 (1.0f / (float)CK);                              // mean L1
    if (b == 0) out[8] = (float)OO;                               // P = 33
  }
}

// ---------------------------------------------------------------------------
extern "C" void kernel_launch(void* const* d_in, const int* in_sizes, int n_in,
                              void* d_out, int out_size, void* d_ws, size_t ws_size,
                              hipStream_t stream) {
  (void)in_sizes; (void)n_in; (void)out_size; (void)ws_size;
  const float* lr  = (const float*)d_in[0];   // [8,3,64,64] f32
  const float* ref = (const float*)d_in[1];   // [8,3,96,96] f32
  float* out  = (float*)d_out;                // [8 idx, 1 P, 8 relevance]
  float* part = (float*)d_ws;                 // [8][1089][8] slice partials

  dim3 grid1(NB, OO * 3);                     // 8 x 99 blocks
  ps_sad<<<grid1, 256, 0, stream>>>(lr, ref, part);
  ps_argmin<<<NB, 256, 0, stream>>>(part, out);
}